// GRAND_52458730553698
// MI455X (gfx1250) — compile-verified
//
#include <hip/hip_runtime.h>

#define DIM 128
#define KHOPS 3
#define NSLOPE 0.2f
#define TB 256

typedef __attribute__((ext_vector_type(2))) float v2f;
typedef __attribute__((ext_vector_type(8))) float v8f;

// ---- monotonic float<->uint key for atomic max over possibly-negative floats
__device__ __forceinline__ unsigned fkey(float f) {
    unsigned u = __float_as_uint(f);
    return (u & 0x80000000u) ? ~u : (u | 0x80000000u);
}
__device__ __forceinline__ float funkey(unsigned k) {
    unsigned u = (k & 0x80000000u) ? (k & 0x7FFFFFFFu) : ~k;
    return __uint_as_float(u);
}

// ---- degree accumulation: deg_in by dst, deg_out by src --------------------
__global__ __launch_bounds__(TB) void k_degree(const int* __restrict__ src,
                                               const int* __restrict__ dst,
                                               float* __restrict__ deg_out,
                                               float* __restrict__ deg_in, int E) {
    int e = blockIdx.x * TB + threadIdx.x;
    if (e >= E) return;
    atomicAdd(&deg_in[dst[e]], 1.0f);
    atomicAdd(&deg_out[src[e]], 1.0f);
}

// ---- deg -> deg^-1/2 in place ---------------------------------------------
__global__ __launch_bounds__(TB) void k_norm(float* __restrict__ n1,
                                             float* __restrict__ n2, int n) {
    int i = blockIdx.x * TB + threadIdx.x;
    if (i >= n) return;
    n1[i] = rsqrtf(fmaxf(n1[i], 1.0f));
    n2[i] = rsqrtf(fmaxf(n2[i], 1.0f));
}

// ---- rst accumulation: rst[dst] += feats[src]*norm2[src] (one wave/edge) ---
__global__ __launch_bounds__(TB) void k_spmm_rst(const int* __restrict__ src,
                                                 const int* __restrict__ dst,
                                                 const float* __restrict__ feats,
                                                 const float* __restrict__ norm2,
                                                 float* __restrict__ rst, int E) {
    int e = (blockIdx.x * TB + threadIdx.x) >> 5;
    int lane = threadIdx.x & 31;
    if (e >= E) return;
    int s = src[e], d = dst[e];
    float sc = norm2[s];
    float4 v = ((const float4*)(feats + (size_t)s * DIM))[lane];
    float* o = rst + (size_t)d * DIM + lane * 4;
    atomicAdd(o + 0, v.x * sc);
    atomicAdd(o + 1, v.y * sc);
    atomicAdd(o + 2, v.z * sc);
    atomicAdd(o + 3, v.w * sc);
}

// ---- rst *= norm1[row]; y = feats (fused copy) -----------------------------
__global__ __launch_bounds__(TB) void k_rstscale_copy(float* __restrict__ rst,
                                                      const float* __restrict__ norm1,
                                                      const float* __restrict__ feats,
                                                      float* __restrict__ y, size_t nd) {
    size_t i = (size_t)blockIdx.x * TB + threadIdx.x;
    if (i >= nd) return;
    rst[i] *= norm1[i >> 7];  // DIM == 128
    y[i] = feats[i];
}

// ---- per-edge weight w = norm1[src] * norm2[dst] (faithful to reference) ---
__global__ __launch_bounds__(TB) void k_wedge(const int* __restrict__ src,
                                              const int* __restrict__ dst,
                                              const float* __restrict__ norm1,
                                              const float* __restrict__ norm2,
                                              float* __restrict__ w, int E) {
    int e = blockIdx.x * TB + threadIdx.x;
    if (e >= E) return;
    w[e] = norm1[src[e]] * norm2[dst[e]];
}

// ---- propagation hop: xout[dst] += xin[src] * w[e] -------------------------
__global__ __launch_bounds__(TB) void k_spmm_hop(const int* __restrict__ src,
                                                 const int* __restrict__ dst,
                                                 const float* __restrict__ xin,
                                                 const float* __restrict__ w,
                                                 float* __restrict__ xout, int E) {
    int e = (blockIdx.x * TB + threadIdx.x) >> 5;
    int lane = threadIdx.x & 31;
    if (e >= E) return;
    int s = src[e], d = dst[e];
    float sc = w[e];
    float4 v = ((const float4*)(xin + (size_t)s * DIM))[lane];
    float* o = xout + (size_t)d * DIM + lane * 4;
    atomicAdd(o + 0, v.x * sc);
    atomicAdd(o + 1, v.y * sc);
    atomicAdd(o + 2, v.z * sc);
    atomicAdd(o + 3, v.w * sc);
}

// ---- y += x ----------------------------------------------------------------
__global__ __launch_bounds__(TB) void k_accum(float* __restrict__ y,
                                              const float* __restrict__ x, size_t nd) {
    size_t i = (size_t)blockIdx.x * TB + threadIdx.x;
    if (i < nd) y[i] += x[i];
}

// ---- h = (y * 0.25) @ W via V_WMMA_F32_16X16X4_F32 -------------------------
// One wave -> one 16x16 output tile. A (16x4 f32): lanes 0-15 hold K=k0,k0+1
// in v[0],v[1]; lanes 16-31 hold K=k0+2,k0+3. B (4x16) mirrored across the
// half-waves. C/D: VGPR i holds row i (lanes 0-15) / row i+8 (lanes 16-31).
__global__ __launch_bounds__(TB) void k_gemm(const float* __restrict__ A,
                                             const float* __restrict__ W,
                                             float* __restrict__ H, int n_rows) {
    int wid = (blockIdx.x * TB + threadIdx.x) >> 5;
    int lane = threadIdx.x & 31;
    const int tiles_n = DIM / 16;  // 8
    int tm = wid / tiles_n;
    int tn = wid - tm * tiles_n;
    if (tm * 16 >= n_rows) return;
    int half = lane >> 4;  // 0: K pair {0,1}, 1: K pair {2,3}
    int l = lane & 15;
    const float* Arow = A + (size_t)(tm * 16 + l) * DIM;
    int bcol = tn * 16 + l;
    v8f acc = {};
    for (int k0 = 0; k0 < DIM; k0 += 4) {
        v2f a, b;
        int ka = k0 + 2 * half;
        a.x = Arow[ka + 0] * 0.25f;  // fold y/(order+1) into the A fragment
        a.y = Arow[ka + 1] * 0.25f;
        b.x = W[(size_t)(ka + 0) * DIM + bcol];
        b.y = W[(size_t)(ka + 1) * DIM + bcol];
        acc = __builtin_amdgcn_wmma_f32_16x16x4_f32(false, a, false, b,
                                                    (short)0, acc, false, false);
    }
    int col = tn * 16 + l;
#pragma unroll
    for (int i = 0; i < 8; ++i) {
        int row = tm * 16 + i + 8 * half;
        H[(size_t)row * DIM + col] = acc[i];
    }
}

// ---- el/er = h . attn_{l,r} (one wave per node) ----------------------------
__global__ __launch_bounds__(TB) void k_attn(const float* __restrict__ H,
                                             const float* __restrict__ al,
                                             const float* __restrict__ ar,
                                             float* __restrict__ el,
                                             float* __restrict__ er, int n) {
    int node = (blockIdx.x * TB + threadIdx.x) >> 5;
    int lane = threadIdx.x & 31;
    if (node >= n) return;
    float4 h4 = ((const float4*)(H + (size_t)node * DIM))[lane];
    float4 l4 = ((const float4*)al)[lane];
    float4 r4 = ((const float4*)ar)[lane];
    float sl = h4.x * l4.x + h4.y * l4.y + h4.z * l4.z + h4.w * l4.w;
    float sr = h4.x * r4.x + h4.y * r4.y + h4.z * r4.z + h4.w * r4.w;
    for (int off = 16; off; off >>= 1) {
        sl += __shfl_xor(sl, off, 32);
        sr += __shfl_xor(sr, off, 32);
    }
    if (lane == 0) { el[node] = sl; er[node] = sr; }
}

// ---- e = leaky_relu(el[src]+er[dst]); per-dst running max (key trick) ------
__global__ __launch_bounds__(TB) void k_edge_e(const int* __restrict__ src,
                                               const int* __restrict__ dst,
                                               const float* __restrict__ el,
                                               const float* __restrict__ er,
                                               float* __restrict__ ebuf,
                                               unsigned* __restrict__ mkey, int E) {
    int e = blockIdx.x * TB + threadIdx.x;
    if (e >= E) return;
    float v = el[src[e]] + er[dst[e]];
    v = v > 0.0f ? v : NSLOPE * v;
    ebuf[e] = v;
    atomicMax(&mkey[dst[e]], fkey(v));
}

// ---- a = exp(e - m[dst]); s[dst] += a (a stored back over e) ---------------
__global__ __launch_bounds__(TB) void k_edge_a(const int* __restrict__ dst,
                                               float* __restrict__ ebuf,
                                               const unsigned* __restrict__ mkey,
                                               float* __restrict__ ssum, int E) {
    int e = blockIdx.x * TB + threadIdx.x;
    if (e >= E) return;
    int d = dst[e];
    float a = expf(ebuf[e] - funkey(mkey[d]));
    ebuf[e] = a;
    atomicAdd(&ssum[d], a);
}

// ---- out[dst] += h[src] * (a[e]/s[dst]) (one wave per edge) ----------------
__global__ __launch_bounds__(TB) void k_spmm_gat(const int* __restrict__ src,
                                                 const int* __restrict__ dst,
                                                 const float* __restrict__ H,
                                                 const float* __restrict__ ebuf,
                                                 const float* __restrict__ ssum,
                                                 float* __restrict__ out, int E) {
    int e = (blockIdx.x * TB + threadIdx.x) >> 5;
    int lane = threadIdx.x & 31;
    if (e >= E) return;
    int s = src[e], d = dst[e];
    float alpha = ebuf[e] / ssum[d];
    float4 v = ((const float4*)(H + (size_t)s * DIM))[lane];
    float* o = out + (size_t)d * DIM + lane * 4;
    atomicAdd(o + 0, v.x * alpha);
    atomicAdd(o + 1, v.y * alpha);
    atomicAdd(o + 2, v.z * alpha);
    atomicAdd(o + 3, v.w * alpha);
}

// ---- out = log_softmax(relu(out + bias)) row-wise (one wave per node) ------
__global__ __launch_bounds__(TB) void k_final(float* __restrict__ out,
                                              const float* __restrict__ bias, int n) {
    int node = (blockIdx.x * TB + threadIdx.x) >> 5;
    int lane = threadIdx.x & 31;
    if (node >= n) return;
    float* row = out + (size_t)node * DIM;
    float v[4];
    float mx = 0.0f;  // relu output >= 0, so 0 is a valid lower bound for max
#pragma unroll
    for (int j = 0; j < 4; ++j) {
        float t = fmaxf(row[lane * 4 + j] + bias[lane * 4 + j], 0.0f);
        v[j] = t;
        mx = fmaxf(mx, t);
    }
    for (int off = 16; off; off >>= 1) mx = fmaxf(mx, __shfl_xor(mx, off, 32));
    float sum = 0.0f;
#pragma unroll
    for (int j = 0; j < 4; ++j) sum += expf(v[j] - mx);
    for (int off = 16; off; off >>= 1) sum += __shfl_xor(sum, off, 32);
    float lse = logf(sum) + mx;
#pragma unroll
    for (int j = 0; j < 4; ++j) row[lane * 4 + j] = v[j] - lse;
}

extern "C" void kernel_launch(void* const* d_in, const int* in_sizes, int n_in,
                              void* d_out, int out_size, void* d_ws, size_t ws_size,
                              hipStream_t stream) {
    const float* feats  = (const float*)d_in[0];
    const float* W      = (const float*)d_in[1];
    const float* attn_l = (const float*)d_in[2];
    const float* attn_r = (const float*)d_in[3];
    const float* bias   = (const float*)d_in[4];
    const int*   src    = (const int*)d_in[5];
    const int*   dst    = (const int*)d_in[6];
    const int N = in_sizes[0] / DIM;  // 50000
    const int E = in_sizes[5];        // 1600000
    float* out = (float*)d_out;

    // workspace carve-out: 6N + 4*N*D + E floats (~110 MB), all L2-resident
    float*    norm1 = (float*)d_ws;            // N  (in-deg^-1/2, by dst)
    float*    norm2 = norm1 + N;               // N  (out-deg^-1/2, by src)
    float*    ssum  = norm2 + N;               // N  softmax denominators
    unsigned* mkey  = (unsigned*)(ssum + N);   // N  softmax max keys
    float*    el    = (float*)(mkey + N);      // N
    float*    er    = el + N;                  // N
    float*    xA    = er + N;                  // N*D  (rst / hop ping)
    float*    xB    = xA + (size_t)N * DIM;    // N*D  (hop pong)
    float*    y     = xB + (size_t)N * DIM;    // N*D  (feature accumulator)
    float*    h     = y + (size_t)N * DIM;     // N*D  (GEMM output)
    float*    ebuf  = h + (size_t)N * DIM;     // E    (w, then e, then a)

    const size_t ND = (size_t)N * DIM;
    int eb  = (E + TB - 1) / TB;
    int ewb = (int)(((size_t)E * 32 + TB - 1) / TB);
    int ndb = (int)((ND + TB - 1) / TB);
    int nwb = (int)(((size_t)N * 32 + TB - 1) / TB);
    int nb  = (N + TB - 1) / TB;

    (void)hipMemsetAsync(norm1, 0, sizeof(float) * 2 * (size_t)N, stream);
    (void)hipMemsetAsync(ssum, 0, sizeof(float) * (size_t)N, stream);
    (void)hipMemsetAsync(mkey, 0, sizeof(unsigned) * (size_t)N, stream);
    (void)hipMemsetAsync(xA, 0, sizeof(float) * ND, stream);
    (void)hipMemsetAsync(out, 0, sizeof(float) * ND, stream);

    // ---- GRAND conv ----
    k_degree<<<eb, TB, 0, stream>>>(src, dst, norm2, norm1, E);
    k_norm<<<nb, TB, 0, stream>>>(norm1, norm2, N);
    k_spmm_rst<<<ewb, TB, 0, stream>>>(src, dst, feats, norm2, xA, E);
    k_rstscale_copy<<<ndb, TB, 0, stream>>>(xA, norm1, feats, y, ND);
    k_wedge<<<eb, TB, 0, stream>>>(src, dst, norm1, norm2, ebuf, E);
    float* xin = xA;
    float* xout = xB;
    for (int k = 0; k < KHOPS; ++k) {
        (void)hipMemsetAsync(xout, 0, sizeof(float) * ND, stream);
        k_spmm_hop<<<ewb, TB, 0, stream>>>(src, dst, xin, ebuf, xout, E);
        k_accum<<<ndb, TB, 0, stream>>>(y, xout, ND);
        float* t = xin; xin = xout; xout = t;
    }

    // ---- GAT ----
    int tiles = (N / 16) * (DIM / 16);
    int gwb = (int)(((size_t)tiles * 32 + TB - 1) / TB);
    k_gemm<<<gwb, TB, 0, stream>>>(y, W, h, N);
    k_attn<<<nwb, TB, 0, stream>>>(h, attn_l, attn_r, el, er, N);
    k_edge_e<<<eb, TB, 0, stream>>>(src, dst, el, er, ebuf, mkey, E);
    k_edge_a<<<eb, TB, 0, stream>>>(dst, ebuf, mkey, ssum, E);
    k_spmm_gat<<<ewb, TB, 0, stream>>>(src, dst, h, ebuf, ssum, out, E);
    k_final<<<nwb, TB, 0, stream>>>(out, bias, N);
}